// NonMaximumSuppress_70540542870216
// MI455X (gfx1250) — compile-verified
//
#include <hip/hip_runtime.h>
#include <cstdint>

#define NANCH 8732
#define NBATCH 2
#define NCONF 21
#define TPB 256

typedef int v4i_t __attribute__((vector_size(4 * sizeof(int))));

__device__ __forceinline__ int imin(int a, int b) { return a < b ? a : b; }

// ---------------------------------------------------------------------------
// Kernel 1: decode boxes + softmax confidences, count valid (score >= 0.5)
// ---------------------------------------------------------------------------
__global__ void __launch_bounds__(TPB) decode_kernel(
    const float* __restrict__ f0, const float* __restrict__ f1,
    const float* __restrict__ f2, const float* __restrict__ f3,
    const float* __restrict__ f4, const float* __restrict__ f5,
    float* __restrict__ box, float* __restrict__ score,
    float* __restrict__ conf, int* __restrict__ vcnt)
{
    int t = blockIdx.x * TPB + threadIdx.x;
    if (t >= NBATCH * NANCH) return;
    int b = t / NANCH, n = t % NANCH;

    const int offs[7] = {0, 5776, 7942, 8542, 8692, 8728, 8732};
    const int Hs[6]   = {38, 19, 10, 5, 3, 1};
    const int As[6]   = {4, 6, 6, 6, 4, 4};
    int k = 0;
#pragma unroll
    for (int q = 0; q < 5; ++q)
        if (n >= offs[q + 1]) k = q + 1;

    int loc = n - offs[k];
    int H = Hs[k], A = As[k];
    int a = loc % A, cell = loc / A;
    int x = cell % H, y = cell / H;

    const double SC[7] = {0.1, 0.2, 0.375, 0.55, 0.725, 0.9, 1.075};
    const double ARt[6][5] = {
        {1.0, 2.0, 0.5, 0.0, 0.0},
        {1.0, 2.0, 0.5, 3.0, 0.3333},
        {1.0, 2.0, 0.5, 3.0, 0.3333},
        {1.0, 2.0, 0.5, 3.0, 0.3333},
        {1.0, 2.0, 0.5, 0.0, 0.0},
        {1.0, 2.0, 0.5, 0.0, 0.0}};
    double s = SC[k], sn = SC[k + 1];
    float dh, dw;
    if (a == A - 1) {                    // extra prior: sqrt(s * s_next)
        float sp = (float)sqrt(s * sn);
        dh = sp; dw = sp;
    } else {
        double r = sqrt(ARt[k][a]);
        dh = (float)(s / r);             // h = s/sqrt(ar)
        dw = (float)(s * r);             // w = s*sqrt(ar)
    }
    float dcx = (x + 0.5f) / (float)H;
    float dcy = (y + 0.5f) / (float)H;

    const float* F[6] = {f0, f1, f2, f3, f4, f5};
    const float* ft = F[k] + ((((size_t)b * H + y) * H + x) * A + a) * (4 + NCONF);

    float t0 = ft[0], t1 = ft[1], t2 = ft[2], t3 = ft[3];

    float l[NCONF];
    float m = -3.4e38f;
#pragma unroll
    for (int c = 0; c < NCONF; ++c) { l[c] = ft[4 + c]; m = fmaxf(m, l[c]); }
    float sum = 0.f;
#pragma unroll
    for (int c = 0; c < NCONF; ++c) { l[c] = expf(l[c] - m); sum += l[c]; }
    float inv = 1.f / sum;
    float cmax = 0.f;
#pragma unroll
    for (int c = 0; c < NCONF; ++c) {
        l[c] *= inv;
        if (c < NCONF - 1) cmax = fmaxf(cmax, l[c]);   // exclude background
    }

    float cx = dcx + dw * t0;
    float cy = dcy + dh * t1;
    float hh = dh * expf(t2);
    float ww = dw * expf(t3);

    size_t idx = (size_t)b * NANCH + n;
    float* bp = box + idx * 4;
    bp[0] = cx; bp[1] = cy; bp[2] = hh; bp[3] = ww;
    score[idx] = cmax;
    float* cp = conf + idx * NCONF;
#pragma unroll
    for (int c = 0; c < NCONF; ++c) cp[c] = l[c];

    if (cmax >= 0.5f) atomicAdd(&vcnt[b], 1);   // order-independent -> deterministic
}

// ---------------------------------------------------------------------------
// Kernel 2: stable descending rank (O(N^2), LDS-tiled) + scatter to sorted
// ---------------------------------------------------------------------------
__global__ void __launch_bounds__(TPB) rank_scatter_kernel(
    const float* __restrict__ score, const float* __restrict__ box,
    const float* __restrict__ conf,
    float* __restrict__ sbox, float* __restrict__ sxyxy,
    float* __restrict__ sscore, float* __restrict__ sconf)
{
    int b = blockIdx.y;
    int i = blockIdx.x * TPB + threadIdx.x;
    size_t bN = (size_t)b * NANCH;
    bool act = i < NANCH;
    float myS = act ? score[bN + i] : 0.f;

    __shared__ float st[TPB];
    int rank = 0;
    for (int t0 = 0; t0 < NANCH; t0 += TPB) {      // uniform trip count
        int j = t0 + threadIdx.x;
        __syncthreads();
        st[threadIdx.x] = (j < NANCH) ? score[bN + j] : -1.f;
        __syncthreads();
        int lim = imin(TPB, NANCH - t0);
        for (int jj = 0; jj < lim; ++jj) {
            float sj = st[jj];
            int jg = t0 + jj;
            // stable: higher score first, ties broken by lower original index
            rank += (sj > myS) || (sj == myS && jg < i);
        }
    }
    if (!act) return;

    size_t r = bN + rank;                          // ranks form a permutation
    sscore[r] = myS;
    const float* bp = box + (bN + i) * 4;
    float cx = bp[0], cy = bp[1], hh = bp[2], ww = bp[3];
    float* sb = sbox + r * 4;
    sb[0] = cx; sb[1] = cy; sb[2] = hh; sb[3] = ww;
    float* sx = sxyxy + r * 4;
    sx[0] = cx - ww * 0.5f; sx[1] = cy - hh * 0.5f;
    sx[2] = cx + ww * 0.5f; sx[3] = cy + hh * 0.5f;
    const float* cp = conf + (bN + i) * NCONF;
    float* scp = sconf + r * NCONF;
#pragma unroll
    for (int c = 0; c < NCONF; ++c) scp[c] = cp[c];
}

// ---------------------------------------------------------------------------
// Kernel 3: suppression mask (valid boxes form a sorted prefix of length V,
// so only i < min(j, V) can suppress) + masked output write.
// IoU tiles staged into LDS via gfx1250 async global->LDS copies (ASYNCcnt).
// ---------------------------------------------------------------------------
__global__ void __launch_bounds__(TPB) nms_out_kernel(
    const float* __restrict__ sbox, const float* __restrict__ sxyxy,
    const float* __restrict__ sconf, const int* __restrict__ vcnt,
    float* __restrict__ out)
{
    int b = blockIdx.y;
    int j = blockIdx.x * TPB + threadIdx.x;
    size_t bN = (size_t)b * NANCH;
    int V = vcnt[b];
    if (V > NANCH) V = NANCH;

    __shared__ __align__(16) float tile[TPB * 4];

    bool havej = j < NANCH;
    bool isval = havej && (j < V);
    float a0 = 0.f, a1 = 0.f, a2 = 0.f, a3 = 0.f, areaA = 0.f;
    if (isval) {
        const float* p = sxyxy + (bN + j) * 4;
        a0 = p[0]; a1 = p[1]; a2 = p[2]; a3 = p[3];
        areaA = (a2 - a0) * (a3 - a1);
    }
    int limit = isval ? j : 0;                 // i in [0, min(j,V)) = [0, j) when j<V

    // Uniform per-block tile loop bound (barrier safety)
    int firstJ = blockIdx.x * TPB;
    int blockLimit = (firstJ >= V) ? 0 : imin(firstJ + TPB - 1, V);

    bool sup = false;
    for (int t0 = 0; t0 < blockLimit; t0 += TPB) {
        __syncthreads();
        int ii = t0 + threadIdx.x;
        if (ii >= NANCH) ii = NANCH - 1;       // clamp (data unused past limit)
        const float* g = sxyxy + (bN + ii) * 4;
#if __has_builtin(__builtin_amdgcn_global_load_async_to_lds_b128)
        // CDNA5 async DMA: 16B/lane global -> LDS, tracked by ASYNCcnt
        __builtin_amdgcn_global_load_async_to_lds_b128(
            (__attribute__((address_space(1))) v4i_t*)g,
            (__attribute__((address_space(3))) v4i_t*)&tile[threadIdx.x * 4],
            0, 0);
#if __has_builtin(__builtin_amdgcn_s_wait_asynccnt)
        __builtin_amdgcn_s_wait_asynccnt(0);
#else
        asm volatile("s_wait_asynccnt 0" ::: "memory");
#endif
#else
        tile[threadIdx.x * 4 + 0] = g[0];
        tile[threadIdx.x * 4 + 1] = g[1];
        tile[threadIdx.x * 4 + 2] = g[2];
        tile[threadIdx.x * 4 + 3] = g[3];
#endif
        __syncthreads();

        int lim = limit - t0;
        if (lim > TPB) lim = TPB;
        for (int kk = 0; kk < lim; ++kk) {
            float b0 = tile[kk * 4 + 0], b1 = tile[kk * 4 + 1];
            float b2 = tile[kk * 4 + 2], b3 = tile[kk * 4 + 3];
            float lx = fmaxf(a0, b0), ly = fmaxf(a1, b1);
            float rx = fminf(a2, b2), ry = fminf(a3, b3);
            float iw = fmaxf(rx - lx, 0.f), ih = fmaxf(ry - ly, 0.f);
            float inter = iw * ih;
            float areaB = (b2 - b0) * (b3 - b1);
            float iou = inter / (areaA + areaB - inter + 1e-8f);
            sup = sup || (iou >= 0.5f);
        }
    }

    if (!havej) return;
    float m = (isval && !sup) ? 1.f : 0.f;
    float* o = out + (bN + j) * (4 + NCONF);
    const float* sb = sbox + (bN + j) * 4;
    o[0] = sb[0] * m; o[1] = sb[1] * m; o[2] = sb[2] * m; o[3] = sb[3] * m;
    const float* sc = sconf + (bN + j) * NCONF;
#pragma unroll
    for (int c = 0; c < NCONF; ++c) o[4 + c] = sc[c] * m;
}

// ---------------------------------------------------------------------------
// Host launcher
// ---------------------------------------------------------------------------
extern "C" void kernel_launch(void* const* d_in, const int* in_sizes, int n_in,
                              void* d_out, int out_size, void* d_ws, size_t ws_size,
                              hipStream_t stream)
{
    (void)in_sizes; (void)n_in; (void)out_size; (void)ws_size;

    const float* f0 = (const float*)d_in[0];
    const float* f1 = (const float*)d_in[1];
    const float* f2 = (const float*)d_in[2];
    const float* f3 = (const float*)d_in[3];
    const float* f4 = (const float*)d_in[4];
    const float* f5 = (const float*)d_in[5];
    float* out = (float*)d_out;

    const size_t BN = (size_t)NBATCH * NANCH;
    float* ws = (float*)d_ws;
    float* box    = ws;                    // BN*4
    float* score  = box    + BN * 4;       // BN
    float* conf   = score  + BN;           // BN*21
    float* sbox   = conf   + BN * NCONF;   // BN*4
    float* sxyxy  = sbox   + BN * 4;       // BN*4
    float* sscore = sxyxy  + BN * 4;       // BN
    float* sconf  = sscore + BN;           // BN*21
    int*   vcnt   = (int*)(sconf + BN * NCONF);   // NBATCH ints (~3.9 MB total)

    (void)hipMemsetAsync(vcnt, 0, NBATCH * sizeof(int), stream);

    int nBlkAll = (int)((BN + TPB - 1) / TPB);
    decode_kernel<<<nBlkAll, TPB, 0, stream>>>(f0, f1, f2, f3, f4, f5,
                                               box, score, conf, vcnt);

    dim3 grid((NANCH + TPB - 1) / TPB, NBATCH);
    rank_scatter_kernel<<<grid, TPB, 0, stream>>>(score, box, conf,
                                                  sbox, sxyxy, sscore, sconf);

    nms_out_kernel<<<grid, TPB, 0, stream>>>(sbox, sxyxy, sconf, vcnt, out);
}